// LocalHolder1D_9826885173695
// MI455X (gfx1250) — compile-verified
//
#include <hip/hip_runtime.h>
#include <math.h>

// Holder-exponent kernel for (B=8, C=64, L=32768) float32 input.
// out[p] = w0'*log2(max3[p]) + w1'*log2(max5[p]) + w2'*log2(max7[p])
// where wk' = wk * log10(2) (folded on the host). The 3-term weighted sum
// runs on the CDNA5 matrix pipe via V_WMMA_F32_16X16X4_F32
// (A = weight rows, B = Y^T columns).

typedef __attribute__((ext_vector_type(2))) float v2f;
typedef __attribute__((ext_vector_type(8))) float v8f;

#define ROW_L 32768   // positions per row (fixed by the reference)
#define VPT   4       // outputs per thread

// Full half-wave exchange: lane l <-> lane l^16 (ds_swizzle SWAPX16).
__device__ __forceinline__ float swap16(float v) {
    return __int_as_float(
        __builtin_amdgcn_ds_swizzle(__float_as_int(v), 0x401F));
}

__global__ __launch_bounds__(256) void
holder1d_wmma_kernel(const float* __restrict__ x, float* __restrict__ out,
                     float w0, float w1, float w2) {
    const int tid  = blockIdx.x * blockDim.x + threadIdx.x;
    const int base = tid * VPT;                 // flat index of first output
    const int p    = base & (ROW_L - 1);        // position within row
    const int rowb = base - p;                  // row start (flat)

    // Load positions p-4 .. p+7 (12 floats, three aligned float4s).
    float s[12];
    if (p >= 4 && p + 8 <= ROW_L) {
        const float4 a = *(const float4*)(x + base - 4);
        const float4 b = *(const float4*)(x + base);
        const float4 c = *(const float4*)(x + base + 4);
        s[0]=a.x; s[1]=a.y; s[2]=a.z; s[3]=a.w;
        s[4]=b.x; s[5]=b.y; s[6]=b.z; s[7]=b.w;
        s[8]=c.x; s[9]=c.y; s[10]=c.z; s[11]=c.w;
    } else {
#pragma unroll
        for (int i = 0; i < 12; ++i) {
            const int q = p - 4 + i;
            s[i] = (q >= 0 && q < ROW_L) ? x[rowb + q] : -INFINITY;
        }
    }

    // Multi-scale maxima + raw v_log_f32 (log2); inputs are in [0.1, 1.0],
    // so no denorm/Inf fixups are required.
    float Y0[VPT], Y1[VPT], Y2[VPT];
#pragma unroll
    for (int j = 0; j < VPT; ++j) {
        const int c0 = 4 + j;
        const float m3 = fmaxf(fmaxf(s[c0 - 1], s[c0]), s[c0 + 1]);
        const float m5 = fmaxf(m3, fmaxf(s[c0 - 2], s[c0 + 2]));
        const float m7 = fmaxf(m5, fmaxf(s[c0 - 3], s[c0 + 3]));
        Y0[j] = __builtin_amdgcn_logf(m3);
        Y1[j] = __builtin_amdgcn_logf(m5);
        Y2[j] = __builtin_amdgcn_logf(m7);
    }

    // ---- Weighted sum on the matrix pipe: D = A(16x4) x B(4x16) + 0 ----
    // A[m][k] = w[k] for all m (w3 = 0). f32 16x4 A layout:
    //   lanes 0-15: vgpr0=K0, vgpr1=K1 ; lanes 16-31: vgpr0=K2, vgpr1=K3.
    const int  lane = threadIdx.x & 31;
    const bool hi   = lane >= 16;

    v2f A;
    A.x = hi ? w2 : w0;
    A.y = hi ? 0.0f : w1;   // K3 weight = 0

    const v8f Cz = {};      // zero accumulator

    float h[VPT];
#pragma unroll
    for (int j = 0; j < VPT; ++j) {
        const float y0 = Y0[j], y1 = Y1[j], y2 = Y2[j];
        const float y0s = swap16(y0);
        const float y1s = swap16(y1);
        const float y2s = swap16(y2);

        // WMMA #1: columns = positions held by lanes 0-15.
        // B vgpr1 on hi lanes is the K3 row: its (finite) garbage is
        // multiplied by A's zero K3 weight, so no select is needed.
        v2f B1;
        B1.x = hi ? y2s : y0;   // hi lane l: Y2 of lane l-16 (K2 slice)
        B1.y = y1;              // K1 slice (lo lanes); K3 garbage (hi) * 0
        // WMMA #2: columns = positions held by lanes 16-31.
        v2f B2;
        B2.x = hi ? y2 : y0s;   // lo lane l: Y0 of lane l+16
        B2.y = y1s;

        const v8f D1 = __builtin_amdgcn_wmma_f32_16x16x4_f32(
            false, A, false, B1, (short)0, Cz, false, false);
        const v8f D2 = __builtin_amdgcn_wmma_f32_16x16x4_f32(
            false, A, false, B2, (short)0, Cz, false, false);

        // All D rows are identical (A rows identical): D vgpr0 at lane l
        // holds holder(column l%16) of that WMMA; pick own-position result.
        h[j] = hi ? D2.s0 : D1.s0;
    }

    float4 o;
    o.x = h[0]; o.y = h[1]; o.z = h[2]; o.w = h[3];
    *(float4*)(out + base) = o;
}

extern "C" void kernel_launch(void* const* d_in, const int* in_sizes, int n_in,
                              void* d_out, int out_size, void* d_ws, size_t ws_size,
                              hipStream_t stream) {
    const float* x = (const float*)d_in[0];
    float* out = (float*)d_out;
    const int N = in_sizes[0];   // 8*64*32768 = 16,777,216

    // Closed-form slope weights (host, double precision):
    // xr[o] = log10((2o+3)/L); w = (3*xr - sum(xr)) / (3*sum(xr^2) - sum(xr)^2)
    // Fold log10(2) in so the device can use raw log2 (v_log_f32).
    const double Ld = (double)ROW_L;
    double xr[3], s1 = 0.0, s2 = 0.0;
    for (int o = 0; o < 3; ++o) {
        xr[o] = log10((2.0 * o + 3.0) / Ld);
        s1 += xr[o];
        s2 += xr[o] * xr[o];
    }
    const double det   = 3.0 * s2 - s1 * s1;
    const double l10_2 = 0.30102999566398119521;  // log10(2)
    const float w0 = (float)(((3.0 * xr[0] - s1) / det) * l10_2);
    const float w1 = (float)(((3.0 * xr[1] - s1) / det) * l10_2);
    const float w2 = (float)(((3.0 * xr[2] - s1) / det) * l10_2);

    const int threads = 256;
    const int blocks  = N / (threads * VPT);   // 16384, exact
    holder1d_wmma_kernel<<<blocks, threads, 0, stream>>>(x, out, w0, w1, w2);
}